// HierarchicalGraphNet_54511724921405
// MI455X (gfx1250) — compile-verified
//
#include <hip/hip_runtime.h>
#include <hip/hip_bf16.h>

// Problem constants from the reference
#define NN 50000
#define EE 400000
#define CC 384
#define PP 524288  // next pow2 >= EE, bitonic sort size
#define ADD_TO_EDGE_SCORE 0.5f

typedef __attribute__((ext_vector_type(2))) float v2f;
typedef __attribute__((ext_vector_type(8))) float v8f;
typedef __attribute__((ext_vector_type(4))) int   v4i;

#if defined(__gfx1250__) && __has_builtin(__builtin_amdgcn_global_load_async_to_lds_b128)
#define HAVE_ASYNC_LDS 1
typedef __attribute__((address_space(1))) v4i gv4i_t;   // global int4
typedef __attribute__((address_space(3))) v4i lv4i_t;   // LDS int4
#else
#define HAVE_ASYNC_LDS 0
#endif

// ---------------------------------------------------------------------------
// Kernel 1: ps = x @ W[:C], pt = x @ W[C:] via V_WMMA_F32_16X16X4_F32.
// One wave computes a 16-row tile; A = 16x4 chunk of x, B = 4x16 with only
// columns 0 (W_s) and 1 (W_t) nonzero. Memory-bound (AI ~2 FLOP/B), so the
// goal is a tight global_load_b64 + ds_load_b64 + v_wmma loop body with no
// EXEC divergence (WMMA requires EXEC all-ones anyway).
// ---------------------------------------------------------------------------
__global__ void k_matvec_wmma(const float* __restrict__ x,
                              const float* __restrict__ W,
                              float* __restrict__ ps,
                              float* __restrict__ pt) {
    __shared__ float w[2 * CC];

    // Stage W (3072 B) into LDS via the CDNA5 async-copy path when available.
#if HAVE_ASYNC_LDS
    if (threadIdx.x < (2 * CC) / 4) {  // 192 x b128
        gv4i_t* gsrc = (gv4i_t*)(uintptr_t)(W + threadIdx.x * 4);
        lv4i_t* ldst = (lv4i_t*)(unsigned)(uintptr_t)(&w[threadIdx.x * 4]);
        __builtin_amdgcn_global_load_async_to_lds_b128(gsrc, ldst,
                                                       /*offset=*/0, /*cpol=*/0);
    }
#if __has_builtin(__builtin_amdgcn_s_wait_asynccnt)
    __builtin_amdgcn_s_wait_asynccnt(0);
#else
    asm volatile("s_wait_asynccnt 0x0" ::: "memory");
#endif
#else
    for (int i = threadIdx.x; i < 2 * CC; i += blockDim.x) w[i] = W[i];
#endif
    __syncthreads();

    const int wave = threadIdx.x >> 5;
    const int lane = threadIdx.x & 31;
    const int tile = blockIdx.x * (blockDim.x >> 5) + wave;
    if (tile * 16 >= NN) return;   // wave-uniform exit: EXEC stays all-ones
    const int r0 = tile * 16;

    const int m     = lane & 15;        // A row within tile / B,D column
    const int khalf = (lane >> 4) << 1; // lanes 0-15 -> K+0/1, lanes 16-31 -> K+2/3
    const bool live = (m < 2);          // only B columns 0 (W_s) and 1 (W_t) nonzero
    const int wbase = (live ? m : 0) * CC;  // clamped LDS base: loads always legal
    const float* xrow = x + (size_t)(r0 + m) * CC;

    v8f acc = {};
    for (int k = 0; k < CC; k += 16) {
        __builtin_prefetch(xrow + k + 128, 0, 1);  // global_prefetch_b8 (speculative)
#pragma unroll
        for (int kk = 0; kk < 16; kk += 4) {
            // A fragment: two consecutive floats of this row (global_load_b64)
            v2f a = *(const v2f*)(xrow + k + kk + khalf);
            // B fragment: unconditional LDS b64 load, then cndmask to zero
            v2f bv = *(const v2f*)(w + wbase + k + kk + khalf);
            v2f b;
            b.x = live ? bv.x : 0.0f;
            b.y = live ? bv.y : 0.0f;
            acc = __builtin_amdgcn_wmma_f32_16x16x4_f32(
                /*neg_a=*/false, a, /*neg_b=*/false, b,
                /*c_mod=*/(short)0, acc, /*reuse_a=*/false, /*reuse_b=*/false);
        }
    }

    // D layout: VGPR j -> lanes 0-15 hold M=j, lanes 16-31 hold M=j+8; N = lane&15
    if (lane == 0)  { for (int j = 0; j < 8; ++j) ps[r0 + j]     = acc[j]; }
    if (lane == 16) { for (int j = 0; j < 8; ++j) ps[r0 + 8 + j] = acc[j]; }
    if (lane == 1)  { for (int j = 0; j < 8; ++j) pt[r0 + j]     = acc[j]; }
    if (lane == 17) { for (int j = 0; j < 8; ++j) pt[r0 + 8 + j] = acc[j]; }
}

// ---------------------------------------------------------------------------
// Ordered-uint encoding for float atomic max
// ---------------------------------------------------------------------------
__device__ __forceinline__ unsigned enc_ord(float f) {
    unsigned u = __float_as_uint(f);
    return (u >> 31) ? ~u : (u | 0x80000000u);
}
__device__ __forceinline__ float dec_ord(unsigned u) {
    if (u == 0u) return 0.0f; // segment with no edges: reference maps -inf -> 0
    return (u & 0x80000000u) ? __uint_as_float(u ^ 0x80000000u)
                             : __uint_as_float(~u);
}

__global__ void k_init_nodes(unsigned* mu, float* ssum, float* cscore) {
    int i = blockIdx.x * blockDim.x + threadIdx.x;
    if (i >= NN) return;
    mu[i] = 0u;
    ssum[i] = 0.0f;
    cscore[i] = 1.0f;
}

__global__ void k_edge_raw(const float* __restrict__ ps, const float* __restrict__ pt,
                           const int* __restrict__ src, const int* __restrict__ dst,
                           const float* __restrict__ b,
                           float* __restrict__ raw, unsigned* __restrict__ mu) {
    int e = blockIdx.x * blockDim.x + threadIdx.x;
    if (e >= EE) return;
    float r = ps[src[e]] + pt[dst[e]] + b[0];
    raw[e] = r;
    atomicMax(&mu[dst[e]], enc_ord(r));
}

__global__ void k_edge_exp(const int* __restrict__ dst,
                           const unsigned* __restrict__ mu,
                           float* __restrict__ raw, float* __restrict__ ssum) {
    int e = blockIdx.x * blockDim.x + threadIdx.x;
    if (e >= EE) return;
    float m = dec_ord(mu[dst[e]]);
    float ex = __expf(raw[e] - m);
    raw[e] = ex;                   // reuse buffer for exp values
    atomicAdd(&ssum[dst[e]], ex);
}

// score + build sort key: ascending sort of (~enc(score) << 32 | idx) gives
// descending score, ties broken by edge index
__global__ void k_edge_score(const int* __restrict__ dst,
                             const float* __restrict__ raw,
                             const float* __restrict__ ssum,
                             float* __restrict__ esc,
                             unsigned long long* __restrict__ key) {
    int e = blockIdx.x * blockDim.x + threadIdx.x;
    if (e >= PP) return;
    if (e < EE) {
        float v = raw[e] / ssum[dst[e]] + ADD_TO_EDGE_SCORE;
        esc[e] = v;
        key[e] = ((unsigned long long)(~enc_ord(v)) << 32) | (unsigned)e;
    } else {
        key[e] = ~0ull; // pad sorts to the end
    }
}

// Bitonic compare-exchange pass (keys fit in 4 MB -> L2-resident on 192 MB L2)
__global__ void k_bitonic(unsigned long long* __restrict__ keys, unsigned j, unsigned k) {
    unsigned i = blockIdx.x * blockDim.x + threadIdx.x;
    unsigned ixj = i ^ j;
    if (ixj > i && i < PP) {
        unsigned long long a = keys[i], b = keys[ixj];
        bool asc = ((i & k) == 0u);
        if ((a > b) == asc) { keys[i] = b; keys[ixj] = a; }
    }
}

// Greedy matching: sequential scan dependency through matched[] -> one lane.
// matched[] lives in LDS (50000 B) so the dependent loads on the critical
// path hit LDS latency instead of L2/HBM. Rank/cumsum folded in.
__global__ void k_greedy(const unsigned long long* __restrict__ keys,
                         const int* __restrict__ src, const int* __restrict__ dst,
                         const float* __restrict__ esc,
                         int* __restrict__ cluster, float* __restrict__ cscore) {
    __shared__ unsigned char lm[NN];
    for (int i = threadIdx.x; i < NN; i += blockDim.x) lm[i] = 0;
    __syncthreads();
    if (threadIdx.x != 0 || blockIdx.x != 0) return;

    int cnt = 0;
    for (int i = 0; i < EE; ++i) {
        unsigned idx = (unsigned)(keys[i] & 0xFFFFFFFFull);
        int s = src[idx], t = dst[idx];
        if (!lm[s] && !lm[t]) {
            cluster[s] = cnt;
            cluster[t] = cnt;
            lm[s] = 1;
            lm[t] = 1;
            cscore[cnt] = esc[idx];
            ++cnt;
        }
    }
    int r = 0;
    for (int i = 0; i < NN; ++i) {
        if (!lm[i]) { cluster[i] = cnt + r; ++r; }
    }
}

__global__ void k_fill_f32(float* p, float v, int n) {
    int i = blockIdx.x * blockDim.x + threadIdx.x;
    if (i < n) p[i] = v;
}
__global__ void k_fill_i32(int* p, int v, int n) {
    int i = blockIdx.x * blockDim.x + threadIdx.x;
    if (i < n) p[i] = v;
}

// new_x[cluster[row]] += x[row]   (f32 atomics at L2)
__global__ void k_scatter_x(const float* __restrict__ x,
                            const int* __restrict__ cluster,
                            float* __restrict__ newx) {
    long long i = (long long)blockIdx.x * blockDim.x + threadIdx.x;
    if (i >= (long long)NN * CC) return;
    int row = (int)(i / CC);
    int col = (int)(i - (long long)row * CC);
    atomicAdd(&newx[(size_t)cluster[row] * CC + col], x[i]);
}

__global__ void k_scale_x(float* __restrict__ newx, const float* __restrict__ cscore) {
    long long i = (long long)blockIdx.x * blockDim.x + threadIdx.x;
    if (i >= (long long)NN * CC) return;
    newx[i] *= cscore[i / CC];
}

// pack mapped edges: primary key mapped[0] (high), secondary mapped[1] (low)
__global__ void k_mapped_keys(const int* __restrict__ src, const int* __restrict__ dst,
                              const int* __restrict__ cluster,
                              unsigned long long* __restrict__ key) {
    int e = blockIdx.x * blockDim.x + threadIdx.x;
    if (e >= PP) return;
    if (e < EE) {
        unsigned m0 = (unsigned)cluster[src[e]];
        unsigned m1 = (unsigned)cluster[dst[e]];
        key[e] = ((unsigned long long)m0 << 32) | m1;
    } else {
        key[e] = ~0ull;
    }
}

__global__ void k_unpack_edges(const unsigned long long* __restrict__ key,
                               int* __restrict__ out_ei) {
    int e = blockIdx.x * blockDim.x + threadIdx.x;
    if (e >= EE) return;
    unsigned long long kk = key[e];
    out_ei[e]      = (int)(kk >> 32);
    out_ei[EE + e] = (int)(kk & 0xFFFFFFFFull);
}

__global__ void k_batch(const int* __restrict__ batch, const int* __restrict__ cluster,
                        int* __restrict__ out_batch) {
    int i = blockIdx.x * blockDim.x + threadIdx.x;
    if (i >= NN) return;
    out_batch[cluster[i]] = batch[i];
}

// ---------------------------------------------------------------------------
// Launch
// ---------------------------------------------------------------------------
static inline size_t align256(size_t x) { return (x + 255) & ~(size_t)255; }

extern "C" void kernel_launch(void* const* d_in, const int* in_sizes, int n_in,
                              void* d_out, int out_size, void* d_ws, size_t ws_size,
                              hipStream_t stream) {
    const float* x     = (const float*)d_in[0];
    const int*   ei    = (const int*)d_in[1];
    const int*   batch = (const int*)d_in[2];
    const float* W     = (const float*)d_in[3];
    const float* b     = (const float*)d_in[4];
    const int* src = ei;
    const int* dst = ei + EE;

    // workspace carve-out (~12.6 MB)
    char* w = (char*)d_ws;
    float*              ps     = (float*)w;              w += align256(NN * 4);
    float*              pt     = (float*)w;              w += align256(NN * 4);
    unsigned*           mu     = (unsigned*)w;           w += align256(NN * 4);
    float*              ssum   = (float*)w;              w += align256(NN * 4);
    float*              raw    = (float*)w;              w += align256(EE * 4);
    float*              esc    = (float*)w;              w += align256(EE * 4);
    float*              cscore = (float*)w;              w += align256(NN * 4);
    unsigned long long* key1   = (unsigned long long*)w; w += align256((size_t)PP * 8);
    unsigned long long* key2   = (unsigned long long*)w; w += align256((size_t)PP * 8);
    (void)ws_size;

    // output regions (concatenated in reference return order)
    float* out_x       = (float*)d_out;
    int*   out_ei      = (int*)(out_x + (size_t)NN * CC);
    int*   out_batch   = out_ei + 2 * EE;
    int*   out_cluster = out_batch + NN;
    (void)out_size; (void)n_in; (void)in_sizes;

    const int B = 256;
    const int gN  = (NN + B - 1) / B;
    const int gE  = (EE + B - 1) / B;
    const int gP  = PP / B;
    const long long NC = (long long)NN * CC;
    const int gNC = (int)((NC + B - 1) / B);

    // 1. projections (WMMA f32)
    {
        int tiles = NN / 16;                    // 3125 (exact)
        int wavesPerBlock = B / 32;             // 8
        int blocks = (tiles + wavesPerBlock - 1) / wavesPerBlock;
        k_matvec_wmma<<<blocks, B, 0, stream>>>(x, W, ps, pt);
    }

    // 2. per-node init
    k_init_nodes<<<gN, B, 0, stream>>>(mu, ssum, cscore);

    // 3. segment softmax
    k_edge_raw  <<<gE, B, 0, stream>>>(ps, pt, src, dst, b, raw, mu);
    k_edge_exp  <<<gE, B, 0, stream>>>(dst, mu, raw, ssum);
    k_edge_score<<<gP, B, 0, stream>>>(dst, raw, ssum, esc, key1);

    // 4. sort edges by descending score (bitonic, L2-resident keys)
    for (unsigned k = 2; k <= PP; k <<= 1)
        for (unsigned j = k >> 1; j >= 1; j >>= 1)
            k_bitonic<<<gP, B, 0, stream>>>(key1, j, k);

    // 5. sequential greedy matching + rank assignment (LDS matched flags)
    k_greedy<<<1, B, 0, stream>>>(key1, src, dst, esc, out_cluster, cscore);

    // 6. new_x = segment_sum(x, cluster) * cscore
    k_fill_f32 <<<gNC, B, 0, stream>>>(out_x, 0.0f, (int)NC);
    k_scatter_x<<<gNC, B, 0, stream>>>(x, out_cluster, out_x);
    k_scale_x  <<<gNC, B, 0, stream>>>(out_x, cscore);

    // 7. remapped edges, lexsorted by (mapped0, mapped1)
    k_mapped_keys<<<gP, B, 0, stream>>>(src, dst, out_cluster, key2);
    for (unsigned k = 2; k <= PP; k <<= 1)
        for (unsigned j = k >> 1; j >= 1; j >>= 1)
            k_bitonic<<<gP, B, 0, stream>>>(key2, j, k);
    k_unpack_edges<<<gE, B, 0, stream>>>(key2, out_ei);

    // 8. new_batch
    k_fill_i32<<<gN, B, 0, stream>>>(out_batch, 0, NN);
    k_batch   <<<gN, B, 0, stream>>>(batch, out_cluster, out_batch);
}